// MHSA_36687610642704
// MI455X (gfx1250) — compile-verified
//
#include <hip/hip_runtime.h>
#include <hip/hip_bf16.h>
#include <math.h>

typedef float v2f __attribute__((ext_vector_type(2)));
typedef float v8f __attribute__((ext_vector_type(8)));

#define NB   32
#define DIM  256
#define TT   120
#define VVN  25
#define HH   8
#define HDm  32
#define SP   (TT*VVN)          /* 3000 spatial columns per n */
#define MF   (3*DIM)           /* 768 fused rows: q | k | v  */
#define NCG  (SP/128 + 1)      /* 24 column groups of 128    */
#define ATT_SCALE 0.17677669529663687f  /* 32^-0.5 */
#define WPB  4                 /* attention waves per block  */

__device__ __forceinline__ v8f wmma4(v2f a, v2f b, v8f c) {
  // V_WMMA_F32_16X16X4_F32: D = A(16x4) x B(4x16) + C(16x16), wave32
  return __builtin_amdgcn_wmma_f32_16x16x4_f32(false, a, false, b, (short)0, c,
                                               false, false);
}

// CDNA5 async global->LDS copy (ASYNCcnt-tracked, no VGPR round trip).
__device__ __forceinline__ void async_g2l_b128(unsigned lds_off, const float* g) {
  asm volatile("global_load_async_to_lds_b128 %0, %1, off"
               :: "v"(lds_off), "v"((unsigned long long)(uintptr_t)g)
               : "memory");
}
#define WAIT_ASYNC_4() asm volatile("s_wait_asynccnt 0x4" ::: "memory")
#define WAIT_ASYNC_0() asm volatile("s_wait_asynccnt 0x0" ::: "memory")

// ---------------------------------------------------------------------------
// Stage A: fused QKV channel GEMM per n:  [768 x 256] @ [256 x 3000]
//   rows   0..255 -> q (W_q) | 256..511 -> k (+e folded) | 512..767 -> v
// Block(256) computes a 64x128 tile; x K-chunks (32x128) are double-buffered
// in LDS via global_load_async_to_lds_b128; each wave does 16x64 via WMMA.
// ---------------------------------------------------------------------------
__global__ __launch_bounds__(256) void qkv_kernel(
    const float* __restrict__ x, const float* __restrict__ e,
    const float* __restrict__ Wkv, const float* __restrict__ Wq,
    float* __restrict__ qb, float* __restrict__ keb, float* __restrict__ vvb)
{
  extern __shared__ float lds[];               // 2 x (32 x 128) f32
  const int tid  = threadIdx.x;
  const int lane = tid & 31, wave = tid >> 5;
  const int n = blockIdx.x;
  const int mw = wave >> 1, nw = wave & 1;
  const int rowBase = blockIdx.y * 64 + mw * 16;
  const int colBase = blockIdx.z * 128;
  const int hl = lane & 15, kh = lane >> 4;

  const float* xn = x + (size_t)n * DIM * SP;
  const int ar = rowBase + hl;                 // A-matrix row for this lane
  const float* wrow = (ar < DIM) ? (Wq + (size_t)ar * DIM)
                                 : (Wkv + (size_t)(ar - DIM) * DIM);

  // per-thread async geometry: 4 b128 ops/chunk (1024 lane-ops block-wide)
  int akr[4], acc4[4];
  #pragma unroll
  for (int j = 0; j < 4; ++j) {
    const int o = tid + j * 256;
    akr[j]  = o >> 5;                          // x row within chunk (0..31)
    acc4[j] = (o & 31) * 4;                    // dword column (16B granules)
  }

  auto issue = [&](int k0, int buf) {
    #pragma unroll
    for (int j = 0; j < 4; ++j) {
      int gcol = colBase + acc4[j];
      if (gcol > SP - 4) gcol = SP - 4;        // keep 16B-aligned, in-bounds
      const float* g = xn + (size_t)(k0 + akr[j]) * SP + gcol;
      const unsigned ldso =
          (unsigned)(uintptr_t)(&lds[buf * 4096 + akr[j] * 128 + acc4[j]]);
      async_g2l_b128(ldso, g);
    }
  };

  const v8f vzero = {0.f,0.f,0.f,0.f,0.f,0.f,0.f,0.f};
  v8f acc[4] = {vzero, vzero, vzero, vzero};

  issue(0, 0);
  for (int kc = 0; kc < 8; ++kc) {
    const int buf = kc & 1;
    if (kc < 7) {
      issue((kc + 1) * 32, buf ^ 1);
      __builtin_prefetch(&wrow[(kc + 1) * 32], 0, 3);
      WAIT_ASYNC_4();                          // own prev chunk done
    } else {
      WAIT_ASYNC_0();
    }
    __syncthreads();                           // all waves' chunks visible

    const float* lb = lds + buf * 4096;
    v2f af[8];
    #pragma unroll
    for (int s2 = 0; s2 < 8; ++s2) {           // hoist A-fragments (weights)
      const int kk = s2 * 4 + kh * 2;
      af[s2].x = wrow[kc * 32 + kk];
      af[s2].y = wrow[kc * 32 + kk + 1];
    }
    #pragma unroll
    for (int s2 = 0; s2 < 8; ++s2) {
      const int kk = s2 * 4 + kh * 2;
      #pragma unroll
      for (int nt = 0; nt < 4; ++nt) {
        const int c = nw * 64 + nt * 16 + hl;
        v2f b;
        b.x = lb[kk * 128 + c];
        b.y = lb[(kk + 1) * 128 + c];
        acc[nt] = wmma4(af[s2], b, acc[nt]);
      }
    }
    __syncthreads();                           // buffer reusable next iter
  }

  #pragma unroll
  for (int nt = 0; nt < 4; ++nt) {
    const int col = colBase + nw * 64 + nt * 16 + hl;
    if (col >= SP) continue;
    const int t = col / VVN, v = col % VVN;
    #pragma unroll
    for (int i = 0; i < 8; ++i) {
      const int d = rowBase + kh * 8 + i;      // C layout: row = (l>>4)*8 + i
      const float val = acc[nt][i];
      if (d < DIM) {
        const int h = d >> 5, hd = d & 31;
        __builtin_nontemporal_store(val,
            &qb[((((size_t)n*TT + t)*HH + h)*VVN + v)*HDm + hd]);
      } else if (d < 2*DIM) {
        const int dk = d - DIM, h = dk >> 5, hd = dk & 31;
        __builtin_nontemporal_store(val + e[((size_t)n*DIM + dk)*SP + col],
            &keb[((((size_t)n*TT + t)*HH + h)*VVN + v)*HDm + hd]);
      } else {
        const int dv = d - 2*DIM, h = dv >> 5, hd = dv & 31;
        __builtin_nontemporal_store(val,
            &vvb[((((size_t)n*TT + t)*HH + h)*VVN + v)*HDm + hd]);
      }
    }
  }
}

// ---------------------------------------------------------------------------
// Stage A2: group bias  bm[n,t,h,m] = sum_c w1[h,c] * e_k[n,t,h,m,c]
// ---------------------------------------------------------------------------
__global__ __launch_bounds__(256) void biasm_kernel(
    const float* __restrict__ e, const float* __restrict__ w1,
    float* __restrict__ bm)
{
  const int idx = blockIdx.x * blockDim.x + threadIdx.x;
  if (idx >= NB*TT*HH*VVN) return;
  const int m = idx % VVN;
  const int h = (idx / VVN) % HH;
  const int t = (idx / (VVN*HH)) % TT;
  const int n =  idx / (VVN*HH*TT);
  const float* ep = e + ((size_t)n*DIM + h*HDm)*SP + t*VVN + m;
  float s = 0.f;
  #pragma unroll
  for (int c = 0; c < HDm; ++c) s += w1[h*HDm + c] * ep[(size_t)c*SP];
  __builtin_nontemporal_store(s, &bm[(((size_t)n*TT + t)*HH + h)*VVN + m]);
}

// ---------------------------------------------------------------------------
// Stage B: attention per (n,t,h). One wave per head-slice.
// ---------------------------------------------------------------------------
__global__ __launch_bounds__(WPB*32) void attn_kernel(
    const float* __restrict__ qb, const float* __restrict__ keb,
    const float* __restrict__ vvb, const float* __restrict__ bm,
    const float* __restrict__ rpe, const float* __restrict__ outer,
    const float* __restrict__ alpha_p, const int* __restrict__ hops,
    float* __restrict__ pre)
{
  __shared__ float smem[WPB][3*1056 + 32];   // sq | ske | svv | sbm (stride 33)
  const int lane = threadIdx.x & 31;
  const int wave = threadIdx.x >> 5;
  const int gid  = blockIdx.x * WPB + wave;  // 30720 = WPB * gridDim exactly
  const int n = gid / (TT*HH);
  const int t = (gid / HH) % TT;
  const int h = gid % HH;

  float* sq  = smem[wave];
  float* ske = sq  + 1056;
  float* svv = ske + 1056;
  float* sbm = svv + 1056;

  const size_t tb = (((size_t)n*TT + t)*HH + h) * (VVN*HDm);  // *800

  for (int i = lane; i < VVN*HDm; i += 32) {
    const int r = i >> 5, c = i & 31;
    sq [r*33 + c] = __builtin_nontemporal_load(&qb [tb + i]);
    ske[r*33 + c] = __builtin_nontemporal_load(&keb[tb + i]);
    svv[r*33 + c] = __builtin_nontemporal_load(&vvb[tb + i]);
  }
  for (int i = lane; i < 7*HDm; i += 32) {    // zero pad rows 25..31
    const int r = VVN + (i >> 5), c = i & 31;
    sq[r*33+c] = 0.f; ske[r*33+c] = 0.f; svv[r*33+c] = 0.f;
  }
  if (lane < VVN)
    sbm[lane] = __builtin_nontemporal_load(&bm[(((size_t)n*TT+t)*HH+h)*VVN + lane]);
  __syncthreads();

  // hop-RPE row bias: lane r handles query row r
  float rq0=0.f, rq1=0.f, rq2=0.f, rq3=0.f, rq4=0.f;
  if (lane < VVN) {
    #pragma unroll
    for (int c = 0; c < HDm; ++c) {
      const float qv = sq[lane*33 + c];
      rq0 += qv * rpe[0*DIM + h*HDm + c];
      rq1 += qv * rpe[1*DIM + h*HDm + c];
      rq2 += qv * rpe[2*DIM + h*HDm + c];
      rq3 += qv * rpe[3*DIM + h*HDm + c];
      rq4 += qv * rpe[4*DIM + h*HDm + c];
    }
  }

  const int hl = lane & 15, kh = lane >> 4;
  const v8f vzero = {0.f,0.f,0.f,0.f,0.f,0.f,0.f,0.f};

  // GEMM1: S(32x32) = q(32x32) @ ke^T(32x32)
  v8f sacc[2][2] = {{vzero, vzero}, {vzero, vzero}};
  #pragma unroll
  for (int k0 = 0; k0 < HDm; k0 += 4) {
    const int kk = k0 + kh * 2;
    #pragma unroll
    for (int mt = 0; mt < 2; ++mt) {
      v2f a;
      a.x = sq[(mt*16 + hl)*33 + kk];
      a.y = sq[(mt*16 + hl)*33 + kk + 1];
      #pragma unroll
      for (int nt = 0; nt < 2; ++nt) {
        v2f b;                                  // B[k][col] = ke[col][k]
        b.x = ske[(nt*16 + hl)*33 + kk];
        b.y = ske[(nt*16 + hl)*33 + kk + 1];
        sacc[mt][nt] = wmma4(a, b, sacc[mt][nt]);
      }
    }
  }
  __syncthreads();

  // dump S into sq's LDS slab (q no longer needed)
  #pragma unroll
  for (int mt = 0; mt < 2; ++mt)
    #pragma unroll
    for (int nt = 0; nt < 2; ++nt)
      #pragma unroll
      for (int i = 0; i < 8; ++i)
        sq[(mt*16 + kh*8 + i)*33 + nt*16 + hl] = sacc[mt][nt][i];
  __syncthreads();

  // softmax + fuse (alpha * softmax + outer): one row per lane
  const float alpha = alpha_p[0];
  const int r = lane;
  if (r < VVN) {
    const int* hrow = hops + r*VVN;
    float mx = -3.4e38f;
    for (int m = 0; m < VVN; ++m) {
      const int hop = hrow[m];
      const float rb = (hop==0) ? rq0 : (hop==1) ? rq1 : (hop==2) ? rq2
                     : (hop==3) ? rq3 : rq4;
      const float val = (sq[r*33 + m] + rb + sbm[m]) * ATT_SCALE;
      sq[r*33 + m] = val;
      mx = fmaxf(mx, val);
    }
    float sum = 0.f;
    for (int m = 0; m < VVN; ++m) {
      const float ev = __expf(sq[r*33 + m] - mx);
      sq[r*33 + m] = ev;
      sum += ev;
    }
    const float inv = alpha / sum;
    const float* orow = outer + ((size_t)h*VVN + r)*VVN;
    for (int m = 0; m < VVN; ++m)
      sq[r*33 + m] = sq[r*33 + m] * inv + orow[m];
    for (int m = VVN; m < 32; ++m) sq[r*33 + m] = 0.f;   // pad K cols
  } else {
    for (int m = 0; m < 32; ++m) sq[r*33 + m] = 0.f;     // pad rows
  }
  __syncthreads();

  // GEMM2: O(32x32) = P(32x32) @ vv(32x32)
  v8f oacc[2][2] = {{vzero, vzero}, {vzero, vzero}};
  #pragma unroll
  for (int k0 = 0; k0 < 32; k0 += 4) {
    const int kk = k0 + kh * 2;
    #pragma unroll
    for (int mt = 0; mt < 2; ++mt) {
      v2f a;
      a.x = sq[(mt*16 + hl)*33 + kk];
      a.y = sq[(mt*16 + hl)*33 + kk + 1];
      #pragma unroll
      for (int nt = 0; nt < 2; ++nt) {
        v2f b;                                  // B[k][col] = vv[k][col]
        b.x = svv[kk*33 + nt*16 + hl];
        b.y = svv[(kk + 1)*33 + nt*16 + hl];
        oacc[mt][nt] = wmma4(a, b, oacc[mt][nt]);
      }
    }
  }

  #pragma unroll
  for (int mt = 0; mt < 2; ++mt)
    #pragma unroll
    for (int nt = 0; nt < 2; ++nt)
      #pragma unroll
      for (int i = 0; i < 8; ++i) {
        const int v = mt*16 + kh*8 + i;
        if (v < VVN)
          __builtin_nontemporal_store(oacc[mt][nt][i],
                                      &pre[tb + v*HDm + nt*16 + hl]);
      }
}

// ---------------------------------------------------------------------------
// Stage C: out[n] = W_proj(256x256) @ pre[n](256x3000) + b_proj
// pre stored as [n,t,h,v,hd]; K index c = h*32+hd is CONTIGUOUS per column
// within a 32-chunk -> async b128 into [col][k] LDS tiles (stride 36).
// ---------------------------------------------------------------------------
__global__ __launch_bounds__(256) void proj_kernel(
    const float* __restrict__ pre, const float* __restrict__ Wp,
    const float* __restrict__ bp, float* __restrict__ out)
{
  extern __shared__ float lds[];               // 2 x (128 x 36) f32
  const int tid  = threadIdx.x;
  const int lane = tid & 31, wave = tid >> 5;
  const int n = blockIdx.x;
  const int mw = wave >> 1, nw = wave & 1;
  const int rowBase = blockIdx.y * 64 + mw * 16;
  const int colBase = blockIdx.z * 128;
  const int hl = lane & 15, kh = lane >> 4;

  const float* wrow = Wp + (size_t)(rowBase + hl) * DIM;

  // per-thread async geometry: j-th op covers column cc[j], k-rows kr4..kr4+3
  int accj[4], akr4[4]; size_t cbase[4];
  #pragma unroll
  for (int j = 0; j < 4; ++j) {
    const int o = tid + j * 256;
    accj[j] = o >> 3;
    akr4[j] = (o & 7) * 4;
    int col = colBase + accj[j];
    if (col > SP - 1) col = SP - 1;
    const int t = col / VVN, v = col % VVN;
    cbase[j] = (((size_t)n*TT + t)*HH) * (VVN*HDm) + (size_t)v*HDm;
  }

  auto issue = [&](int kc, int buf) {
    #pragma unroll
    for (int j = 0; j < 4; ++j) {
      const float* g = pre + cbase[j] + (size_t)kc * (VVN*HDm) + akr4[j];
      const unsigned ldso =
          (unsigned)(uintptr_t)(&lds[buf * 4608 + accj[j] * 36 + akr4[j]]);
      async_g2l_b128(ldso, g);
    }
  };

  const v8f vzero = {0.f,0.f,0.f,0.f,0.f,0.f,0.f,0.f};
  v8f acc[4] = {vzero, vzero, vzero, vzero};

  issue(0, 0);
  for (int kc = 0; kc < 8; ++kc) {             // h = kc is the head index
    const int buf = kc & 1;
    if (kc < 7) {
      issue(kc + 1, buf ^ 1);
      __builtin_prefetch(&wrow[(kc + 1) * 32], 0, 3);
      WAIT_ASYNC_4();
    } else {
      WAIT_ASYNC_0();
    }
    __syncthreads();

    const float* lb = lds + buf * 4608;
    v2f af[8];
    #pragma unroll
    for (int s2 = 0; s2 < 8; ++s2) {
      const int kk = s2 * 4 + kh * 2;
      af[s2].x = wrow[kc * 32 + kk];
      af[s2].y = wrow[kc * 32 + kk + 1];
    }
    #pragma unroll
    for (int s2 = 0; s2 < 8; ++s2) {
      const int kk = s2 * 4 + kh * 2;
      #pragma unroll
      for (int nt = 0; nt < 4; ++nt) {
        const int c = nw * 64 + nt * 16 + hl;
        v2f b;
        b.x = lb[c * 36 + kk];
        b.y = lb[c * 36 + kk + 1];
        acc[nt] = wmma4(af[s2], b, acc[nt]);
      }
    }
    __syncthreads();
  }

  #pragma unroll
  for (int nt = 0; nt < 4; ++nt) {
    const int col = colBase + nw * 64 + nt * 16 + hl;
    if (col >= SP) continue;
    #pragma unroll
    for (int i = 0; i < 8; ++i) {
      const int d = rowBase + kh * 8 + i;
      __builtin_nontemporal_store(acc[nt][i] + bp[d],
                                  &out[((size_t)n*DIM + d)*SP + col]);
    }
  }
}

// ---------------------------------------------------------------------------
extern "C" void kernel_launch(void* const* d_in, const int* in_sizes, int n_in,
                              void* d_out, int out_size, void* d_ws, size_t ws_size,
                              hipStream_t stream) {
  const float* x     = (const float*)d_in[0];
  const float* e     = (const float*)d_in[1];
  const float* Wkv   = (const float*)d_in[2];
  const float* Wq    = (const float*)d_in[3];
  const float* Wp    = (const float*)d_in[4];
  const float* bp    = (const float*)d_in[5];
  const float* rpe   = (const float*)d_in[6];
  const float* w1    = (const float*)d_in[7];
  const float* outer = (const float*)d_in[8];
  const float* alpha = (const float*)d_in[9];
  const int*   hops  = (const int*)d_in[10];
  float* out = (float*)d_out;

  const size_t SZ = (size_t)NB*TT*HH*VVN*HDm;   // 24,576,000 floats
  float* qb   = (float*)d_ws;
  float* keb  = qb  + SZ;
  float* vvb  = keb + SZ;
  float* preb = vvb + SZ;
  float* bmb  = preb + SZ;                      // + 768,000 floats

  qkv_kernel<<<dim3(NB, MF/64, NCG), 256, 2*32*128*4, stream>>>(
      x, e, Wkv, Wq, qb, keb, vvb);
  biasm_kernel<<<(NB*TT*HH*VVN + 255)/256, 256, 0, stream>>>(e, w1, bmb);
  attn_kernel<<<(NB*TT*HH)/WPB, WPB*32, 0, stream>>>(
      qb, keb, vvb, bmb, rpe, outer, alpha, hops, preb);
  proj_kernel<<<dim3(NB, DIM/64, NCG), 256, 2*128*36*4, stream>>>(
      preb, Wp, bp, out);
}